// WESA_43447889166364
// MI455X (gfx1250) — compile-verified
//
#include <hip/hip_runtime.h>
#include <hip/hip_bf16.h>
#include <math.h>

// ---------------------------------------------------------------------------
// WESA: weighted-ensemble simulated annealing.
//   y_totals : [8, 32, 512, 64] f32   (d_in[0], 8,388,608 floats, 32 MB)
//   l_total  : [32, 512] int64        (d_in[1], 16384 elements)
//   weight   : [8] f32                (d_in[2])
//   output   : w_best [8] f32
//
// Eval = wT * Y matvec + argmax + count; bandwidth-bound, L2-resident (32MB
// << 192MB L2). WMMA f32 16x16x4 performs the 8-model weighted reduce with
// zero redundant loads. 301 evals + 301 scalar SA-update kernels, all state
// in d_ws, integer-atomic counting for determinism under graph replay.
// ---------------------------------------------------------------------------

#define M_MODELS 8
#define N_POS    16384          // 32*512
#define C_CLS    64
#define M_STRIDE (N_POS * C_CLS) // 1,048,576 floats between model planes
#define N_STEPS  300
#define SA_SCALE 0.1f
#define SA_ALPHA 0.98f
#define SA_T0    100.0f
#define SA_TE    1.0f

typedef __attribute__((ext_vector_type(2))) float v2f;
typedef __attribute__((ext_vector_type(8))) float v8f;

// Workspace layout (d_ws):
//   floats  [0..7] w_cur | [8..15] w_best | [16..23] w_prop | [24] loss | [25] T
//   uints @ +64 floats: [0] correct-counter | [1] done flag
#define WSF_CUR   0
#define WSF_BEST  8
#define WSF_PROP  16
#define WSF_LOSS  24
#define WSF_T     25

// ------------------------- threefry2x32 RNG --------------------------------
__device__ __forceinline__ void tf_round(unsigned& x0, unsigned& x1, int rot) {
    x0 += x1;
    x1 = (x1 << rot) | (x1 >> (32 - rot));
    x1 ^= x0;
}

__device__ __forceinline__ uint2 threefry2x32(uint2 key, uint2 ctr) {
    unsigned ks0 = key.x, ks1 = key.y, ks2 = ks0 ^ ks1 ^ 0x1BD11BDAu;
    unsigned x0 = ctr.x + ks0, x1 = ctr.y + ks1;
    tf_round(x0, x1, 13); tf_round(x0, x1, 15); tf_round(x0, x1, 26); tf_round(x0, x1, 6);
    x0 += ks1; x1 += ks2 + 1u;
    tf_round(x0, x1, 17); tf_round(x0, x1, 29); tf_round(x0, x1, 16); tf_round(x0, x1, 24);
    x0 += ks2; x1 += ks0 + 2u;
    tf_round(x0, x1, 13); tf_round(x0, x1, 15); tf_round(x0, x1, 26); tf_round(x0, x1, 6);
    x0 += ks0; x1 += ks1 + 3u;
    tf_round(x0, x1, 17); tf_round(x0, x1, 29); tf_round(x0, x1, 16); tf_round(x0, x1, 24);
    x0 += ks1; x1 += ks2 + 4u;
    tf_round(x0, x1, 13); tf_round(x0, x1, 15); tf_round(x0, x1, 26); tf_round(x0, x1, 6);
    x0 += ks2; x1 += ks0 + 5u;
    return make_uint2(x0, x1);
}

__device__ __forceinline__ float u01(unsigned b) {
    return (float)(b >> 8) * (1.0f / 16777216.0f);   // [0,1)
}

__device__ __forceinline__ float rng_normal(unsigned stream, unsigned slot) {
    uint2 r = threefry2x32(make_uint2(0u, 1u), make_uint2(stream, slot));
    float u1 = fmaxf(u01(r.x), 1.0e-7f);
    float u2 = u01(r.y);
    return sqrtf(-2.0f * logf(u1)) * cosf(6.28318530718f * u2);
}

__device__ __forceinline__ float rng_uniform(unsigned stream, unsigned slot) {
    uint2 r = threefry2x32(make_uint2(2u, 3u), make_uint2(stream, slot));
    return u01(r.x);
}

// ------------------------------ init ---------------------------------------
__global__ void wesa_init(const float* __restrict__ w0,
                          float* __restrict__ wsf,
                          unsigned* __restrict__ wsu) {
    int i = threadIdx.x;
    if (i < M_MODELS) {
        float w = w0[i];
        wsf[WSF_CUR + i]  = w;
        wsf[WSF_BEST + i] = w;
        wsf[WSF_PROP + i] = w;   // first eval is loss0 with the initial weight
    }
    if (i == 0) {
        wsf[WSF_LOSS] = 0.0f;
        wsf[WSF_T]    = SA_T0;
        wsu[0] = 0u;             // correct counter
        wsu[1] = 0u;             // done flag
    }
}

// ------------------------------ eval ---------------------------------------
// One wave per n-row (strided). For each n:
//   4 iterations x [2x V_WMMA_F32_16X16X4_F32] compute S[c] = sum_m w[m]*Y[m,n,c]
//   for 16 consecutive c per iteration. B carries 16 positions x 4 models per
//   WMMA (each lane loads exactly 2 floats per WMMA -> no redundant traffic).
//   D VGPR0: lane L holds S[c = it*16 + (L&15)] (replicated in both halves).
__global__ void __launch_bounds__(256)
wesa_eval(const float* __restrict__ Y,
          const long long* __restrict__ Tgt,
          const float* __restrict__ wprop,
          unsigned* __restrict__ counter) {
    const int lane  = threadIdx.x & 31;
    const int half  = lane >> 4;          // 0: K rows {0,1}, 1: K rows {2,3}
    const int l16   = lane & 15;
    const int wavesPerBlock = blockDim.x >> 5;
    const int gwave  = blockIdx.x * wavesPerBlock + (threadIdx.x >> 5);
    const int nwaves = gridDim.x * wavesPerBlock;

    // A = weights replicated down 16 rows.
    // A-layout (16x4 f32): VGPR0 = K{0|2}, VGPR1 = K{1|3} by half-wave.
    v2f aLo, aHi;
    aLo.x = wprop[half * 2 + 0];
    aLo.y = wprop[half * 2 + 1];
    aHi.x = wprop[4 + half * 2 + 0];
    aHi.y = wprop[4 + half * 2 + 1];

    unsigned correct = 0;
    for (int n = gwave; n < N_POS; n += nwaves) {
        const int base = n * C_CLS;
        // Pull the next row this wave will touch while WMMAs retire.
        __builtin_prefetch(Y + (size_t)(base + nwaves * C_CLS) + lane, 0, 3);

        const long long tgt = Tgt[n];
        float smax = -INFINITY;
        int   cmax = 0;

        #pragma unroll
        for (int it = 0; it < 4; ++it) {
            const int pos = base + it * 16 + l16;
            const float* p = Y + pos;
            const int m0 = half * 2;              // model rows for this half
            // B-layout (4x16 f32), mirroring A: VGPR0 = K{0|2}, VGPR1 = K{1|3}.
            v2f bLo, bHi;
            bLo.x = p[(size_t)(m0 + 0) * M_STRIDE];
            bLo.y = p[(size_t)(m0 + 1) * M_STRIDE];
            bHi.x = p[(size_t)(m0 + 4) * M_STRIDE];
            bHi.y = p[(size_t)(m0 + 5) * M_STRIDE];

            v8f acc = {};
            acc = __builtin_amdgcn_wmma_f32_16x16x4_f32(
                false, aLo, false, bLo, (short)0, acc, false, false);
            acc = __builtin_amdgcn_wmma_f32_16x16x4_f32(
                false, aHi, false, bHi, (short)0, acc, false, false);

            const float s = acc[0];               // S[c] for c = it*16 + l16
            const int   c = it * 16 + l16;
            const bool  b = s > smax;             // strict > keeps first max
            smax = b ? s : smax;
            cmax = b ? c : cmax;
        }

        // Cross-lane argmax, ties -> lowest c (matches jnp.argmax).
        #pragma unroll
        for (int off = 16; off > 0; off >>= 1) {
            float so = __shfl_xor(smax, off, 32);
            int   co = __shfl_xor(cmax, off, 32);
            const bool take = (so > smax) || (so == smax && co < cmax);
            smax = take ? so : smax;
            cmax = take ? co : cmax;
        }
        correct += (lane == 0 && cmax == (int)tgt) ? 1u : 0u;
    }
    if (lane == 0 && correct) atomicAdd(counter, correct);
}

// ------------------------------ update -------------------------------------
// Scalar SA step: read count -> loss_new, accept/reject, anneal T on accept
// only, latch done at T<=TE, then generate the next proposal and publish
// w_best. step==0 just installs loss0.
__global__ void wesa_update(float* __restrict__ wsf,
                            unsigned* __restrict__ wsu,
                            float* __restrict__ out,
                            int step) {
    if (threadIdx.x != 0) return;

    const unsigned cnt = wsu[0];
    wsu[0] = 0u;
    const float loss_new = -(float)cnt / (float)N_POS;

    float loss = wsf[WSF_LOSS];
    float Tcur = wsf[WSF_T];
    const bool done = (wsu[1] != 0u);

    if (step == 0) {
        wsf[WSF_LOSS] = loss_new;
    } else {
        const float u     = rng_uniform((unsigned)step, 0u);
        const bool better = loss_new < loss;
        const bool accept = better || (u < expf((loss - loss_new) / Tcur));
        const bool upd    = accept && !done;
        if (better && !done) {
            #pragma unroll
            for (int m = 0; m < M_MODELS; ++m) wsf[WSF_BEST + m] = wsf[WSF_PROP + m];
        }
        if (upd) {
            #pragma unroll
            for (int m = 0; m < M_MODELS; ++m) wsf[WSF_CUR + m] = wsf[WSF_PROP + m];
            wsf[WSF_LOSS] = loss_new;
            Tcur *= SA_ALPHA;
            wsf[WSF_T] = Tcur;
        }
        if (Tcur <= SA_TE) wsu[1] = 1u;
    }

    // Proposal for step+1: |w + scale*randn|, renormalized to sum 1.
    float wn[M_MODELS];
    float s = 0.0f;
    #pragma unroll
    for (int m = 0; m < M_MODELS; ++m) {
        wn[m] = fabsf(wsf[WSF_CUR + m] + SA_SCALE * rng_normal((unsigned)(step + 1), (unsigned)m));
        s += wn[m];
    }
    const float inv = 1.0f / s;
    #pragma unroll
    for (int m = 0; m < M_MODELS; ++m) wsf[WSF_PROP + m] = wn[m] * inv;

    #pragma unroll
    for (int m = 0; m < M_MODELS; ++m) out[m] = wsf[WSF_BEST + m];
}

// ---------------------------------------------------------------------------
extern "C" void kernel_launch(void* const* d_in, const int* in_sizes, int n_in,
                              void* d_out, int out_size, void* d_ws, size_t ws_size,
                              hipStream_t stream) {
    const float*     Y   = (const float*)d_in[0];
    const long long* Tgt = (const long long*)d_in[1];
    const float*     w0  = (const float*)d_in[2];
    float*    out = (float*)d_out;
    float*    wsf = (float*)d_ws;
    unsigned* wsu = (unsigned*)d_ws + 64;   // +256B, past the float state

    (void)in_sizes; (void)n_in; (void)out_size; (void)ws_size;

    const int EVAL_BLOCKS  = 256;  // 2048 waves -> 8 rows each, L2-streaming
    const int EVAL_THREADS = 256;  // 8 wave32 per block

    wesa_init<<<1, 64, 0, stream>>>(w0, wsf, wsu);

    // loss0 with the initial weight
    wesa_eval<<<EVAL_BLOCKS, EVAL_THREADS, 0, stream>>>(Y, Tgt, wsf + WSF_PROP, wsu);
    wesa_update<<<1, 32, 0, stream>>>(wsf, wsu, out, 0);

    // 300 sequential SA steps
    for (int s = 1; s <= N_STEPS; ++s) {
        wesa_eval<<<EVAL_BLOCKS, EVAL_THREADS, 0, stream>>>(Y, Tgt, wsf + WSF_PROP, wsu);
        wesa_update<<<1, 32, 0, stream>>>(wsf, wsu, out, s);
    }
}